// QuantumConv2d_52836687675707
// MI455X (gfx1250) — compile-verified
//
#include <hip/hip_runtime.h>
#include <stdint.h>

// Problem constants (match reference)
#define N_   16
#define IC_  8
#define OC_  8
#define H_   256
#define W_   256
#define K_   9      // 3x3

// Tiling
#define TW       64            // output tile width per block
#define TH       16            // output tile height per block
#define PITCH    66            // TW + 2 halo
#define HROWS    18            // TH + 2 halo
#define CH_STRIDE (PITCH*HROWS)       // 1188 floats per channel
#define TILE_ELEMS (IC_*CH_STRIDE)    // 9504 floats = 38 KB LDS

// ---------------------------------------------------------------------------
// Main kernel (placed first so the disasm snippet shows it): one block =
// 64x16 output tile, one n, all 8 output channels. 256 threads, each owning
// 4 contiguous output pixels. Input tile (8ch x 18 x 66 with halo) staged
// via per-lane async global->LDS copies (CDNA5 path, tracked by ASYNCcnt).
// ---------------------------------------------------------------------------
__global__ __launch_bounds__(256)
void quanv_kernel(const float* __restrict__ batches,
                  const float* __restrict__ thp,
                  float* __restrict__ out) {
  __shared__ float tile[TILE_ELEMS];

  const int tid = threadIdx.x;
  const int bx = blockIdx.x, by = blockIdx.y, n = blockIdx.z;
  const float* src = batches + (size_t)n * (IC_ * H_ * W_);

  // Border blocks pre-zero the tile so zero-padding falls out naturally
  // (cos(theta*0) == 1, matching the reference's jnp.pad).
  const bool border = (bx == 0) | (by == 0) |
                      (bx == (int)gridDim.x - 1) | (by == (int)gridDim.y - 1);
  if (border) {
    for (int e = tid; e < TILE_ELEMS; e += 256) tile[e] = 0.0f;
  }
  __syncthreads();

  // Async stage the halo tile. GVS mode: SGPR base + signed 32-bit VGPR byte
  // offset. OOB lanes are EXEC-masked off; pre-zeroed LDS supplies padding.
  // Index math is incremental: 256 = 3*66 + 58, so each step carries at most
  // once into r and once into ch (no per-iteration divides).
  {
    const int gx0 = bx * TW - 1;
    const int gy0 = by * TH - 1;
    int r  = tid / PITCH;            // constant divisor -> umulhi sequence
    int c  = tid - r * PITCH;
    int ch = 0;
    for (int e = tid; e < TILE_ELEMS; e += 256) {
      const int gy = gy0 + r;
      const int gx = gx0 + c;
      if ((unsigned)gy < (unsigned)H_ && (unsigned)gx < (unsigned)W_) {
        const int goff = (ch << 20) + (gy << 10) + (gx << 2); // byte offset
        const unsigned laddr = (unsigned)(uintptr_t)(tile + e);
        asm volatile("global_load_async_to_lds_b32 %0, %1, %2"
                     :: "v"(laddr), "v"(goff), "s"(src) : "memory");
      }
      c += 58; r += 3;
      if (c >= PITCH) { c -= PITCH; r += 1; }
      if (r >= HROWS) { r -= HROWS; ch += 1; }
    }
  }
  asm volatile("s_wait_asynccnt 0" ::: "memory");
  __syncthreads();

  // Per-thread pixel coordinates: 16 threads x 4px wide, 16 thread rows.
  const int tx4 = (tid & 15) * 4;   // 0..60, 16B-aligned column within tile
  const int ty  = tid >> 4;         // 0..15

  float acc[OC_][4];
  #pragma unroll
  for (int o = 0; o < OC_; ++o) {
    acc[o][0] = 0.f; acc[o][1] = 0.f; acc[o][2] = 0.f; acc[o][3] = 0.f;
  }

  for (int i = 0; i < IC_; ++i) {
    // 3 rows x 6 cols of the padded input cover all taps for 4 pixels.
    float xv[3][6];
    const float* tch = tile + i * CH_STRIDE;
    #pragma unroll
    for (int dy = 0; dy < 3; ++dy) {
      #pragma unroll
      for (int c = 0; c < 6; ++c) {
        xv[dy][c] = tch[(ty + dy) * PITCH + tx4 + c];
      }
    }
    #pragma unroll
    for (int o = 0; o < OC_; ++o) {
      const float* tp = thp + (size_t)(o * IC_ + i) * K_;  // uniform -> SMEM
      // Tap 0 initializes the product chain (saves the p=1 multiply).
      const float tv0 = tp[0];
      float p0 = __builtin_amdgcn_cosf(tv0 * xv[0][0]);
      float p1 = __builtin_amdgcn_cosf(tv0 * xv[0][1]);
      float p2 = __builtin_amdgcn_cosf(tv0 * xv[0][2]);
      float p3 = __builtin_amdgcn_cosf(tv0 * xv[0][3]);
      #pragma unroll
      for (int k = 1; k < 9; ++k) {
        const int dy = k / 3, dx = k - (k / 3) * 3;        // constants
        const float tv = tp[k];                            // 1/(2*pi)-scaled
        p0 *= __builtin_amdgcn_cosf(tv * xv[dy][dx + 0]);
        p1 *= __builtin_amdgcn_cosf(tv * xv[dy][dx + 1]);
        p2 *= __builtin_amdgcn_cosf(tv * xv[dy][dx + 2]);
        p3 *= __builtin_amdgcn_cosf(tv * xv[dy][dx + 3]);
      }
      acc[o][0] += p0; acc[o][1] += p1; acc[o][2] += p2; acc[o][3] += p3;
    }
  }

  // Store: 8 x float4 per thread, fully coalesced, 16B-aligned.
  const int oy = by * TH + ty;
  const int ox = bx * TW + tx4;
  #pragma unroll
  for (int o = 0; o < OC_; ++o) {
    size_t idx = ((((size_t)n * OC_ + o) * H_ + oy) * W_) + ox;
    float4 v = make_float4(acc[o][0], acc[o][1], acc[o][2], acc[o][3]);
    *(float4*)(out + idx) = v;
  }
}

// ---------------------------------------------------------------------------
// Kernel 1: fold the 1/(2*pi) prescale into theta once (v_cos_f32 computes
// cos(2*pi*x)); saves one VALU mul per cosine in the hot loop (604M of them).
// ---------------------------------------------------------------------------
__global__ void theta_prescale_kernel(const float* __restrict__ theta,
                                      float* __restrict__ thp, int n) {
  int i = blockIdx.x * blockDim.x + threadIdx.x;
  if (i < n) thp[i] = theta[i] * 0.15915494309189535f; // 1/(2*pi)
}

// ---------------------------------------------------------------------------
// Launcher
// ---------------------------------------------------------------------------
extern "C" void kernel_launch(void* const* d_in, const int* in_sizes, int n_in,
                              void* d_out, int out_size, void* d_ws, size_t ws_size,
                              hipStream_t stream) {
  const float* batches = (const float*)d_in[0];   // [16,8,256,256] f32
  const float* theta   = (const float*)d_in[1];   // [64,9] f32
  float* out = (float*)d_out;                     // [16,8,256,256] f32
  float* thp = (float*)d_ws;                      // 576 floats of scratch

  const int nth = OC_ * IC_ * K_;  // 576
  theta_prescale_kernel<<<(nth + 255) / 256, 256, 0, stream>>>(theta, thp, nth);

  dim3 grid(W_ / TW, H_ / TH, N_);   // 4 x 16 x 16 = 1024 blocks
  quanv_kernel<<<grid, 256, 0, stream>>>(batches, thp, out);
}